// Node_Embedding_28896539967495
// MI455X (gfx1250) — compile-verified
//
#include <hip/hip_runtime.h>
#include <hip/hip_bf16.h>

#define N_NODE 50000   // N_DRUG == N_DIS
#define DIM    128
#define NEDGE  500000

typedef float v2f __attribute__((ext_vector_type(2)));
typedef float v4f __attribute__((ext_vector_type(4)));
typedef float v8f __attribute__((ext_vector_type(8)));

// ---------------------------------------------------------------- utilities
__global__ void zero_u32_kernel(unsigned* __restrict__ p, int n) {
    int i = blockIdx.x * blockDim.x + threadIdx.x;
    if (i < n) p[i] = 0u;
}

// degOut[r][node], degIn[r][node], r = blockIdx.y in {0..3}
__global__ void count_degrees_kernel(const int* __restrict__ s0, const int* __restrict__ d0,
                                     const int* __restrict__ s1, const int* __restrict__ d1,
                                     const int* __restrict__ s2, const int* __restrict__ d2,
                                     const int* __restrict__ s3, const int* __restrict__ d3,
                                     unsigned* __restrict__ degOut,
                                     unsigned* __restrict__ degIn) {
    int e = blockIdx.x * blockDim.x + threadIdx.x;
    int r = blockIdx.y;
    if (e >= NEDGE) return;
    const int* sp; const int* dp;
    switch (r) {
        case 0:  sp = s0; dp = d0; break;
        case 1:  sp = s1; dp = d1; break;
        case 2:  sp = s2; dp = d2; break;
        default: sp = s3; dp = d3; break;
    }
    atomicAdd(&degOut[r * N_NODE + sp[e]], 1u);
    atomicAdd(&degIn [r * N_NODE + dp[e]], 1u);
}

// d_out init: drug half gets b[0]+b[2], dis half gets b[1]+b[3]
__global__ void init_out_bias_kernel(const float* __restrict__ b, float* __restrict__ out) {
    int i = blockIdx.x * blockDim.x + threadIdx.x;
    if (i >= 2 * N_NODE * DIM) return;
    int col = i & (DIM - 1);
    float v;
    if (i < N_NODE * DIM) v = b[0 * DIM + col] + b[2 * DIM + col];
    else                  v = b[1 * DIM + col] + b[3 * DIM + col];
    out[i] = v;
}

// ---------------------------------------------------------------- WMMA GEMM
// H[rel] = rowscale(X_src, degOut[rel]^-1/2) @ W[rel]
// grid: (ceil(M/64), DIM/16, 4 relations); block: 1 wave (32 threads)
//
// K-slot relabeling: WMMA accumulation over K is order-independent, so A and B
// may agree on any permutation of logical-k -> instruction K-slot. We use, for
// step t (m = t>>1, odd = t&1):
//   slots (K0,K1) = {8m+2*odd, 8m+2*odd+1}     (lanes 0-15)
//   slots (K2,K3) = {8m+4+2*odd, 8m+4+2*odd+1} (lanes 16-31)
// so a lane's A data for a step-pair is one contiguous float4 (b128 load),
// and all B fragments are preloaded into registers before the loop.
__global__ void __launch_bounds__(32)
gemm_norm_wmma_kernel(const float* __restrict__ x_drug,
                      const float* __restrict__ x_dis,
                      const float* __restrict__ Wt,       // [4,128,128]
                      const unsigned* __restrict__ degOut,
                      float* __restrict__ H) {            // [4, N_NODE, 128]
    const int lane = threadIdx.x;
    const int half = lane >> 4;          // 0: lanes 0-15, 1: lanes 16-31
    const int lm   = lane & 15;
    const int rel  = blockIdx.z;
    const int row0 = blockIdx.x * 64;
    const int col0 = blockIdx.y * 16;

    const float*    X  = (rel < 2) ? x_drug : x_dis;
    const float*    Wr = Wt + (size_t)rel * DIM * DIM;
    const unsigned* dg = degOut + (size_t)rel * N_NODE;

    // Per-subtile A row index + out-degree scale (0 for tail rows => zero rows)
    int   rr[4];
    float sc[4];
#pragma unroll
    for (int s = 0; s < 4; ++s) {
        int r      = row0 + s * 16 + lm;
        bool valid = (r < N_NODE);
        rr[s]      = valid ? r : (N_NODE - 1);
        float d    = (float)dg[rr[s]];
        sc[s]      = valid ? rsqrtf(fmaxf(d, 1.0f)) : 0.0f;
    }

    // Preload all 32 B fragments (relabeled K-slot order) into registers.
    v2f bf[32];
#pragma unroll
    for (int t = 0; t < 32; ++t) {
        int kb = 8 * (t >> 1) + 4 * half + 2 * (t & 1);
        bf[t].x = Wr[(kb    ) * DIM + col0 + lm];
        bf[t].y = Wr[(kb + 1) * DIM + col0 + lm];
    }

    v8f acc[4] = {};
#pragma unroll
    for (int m = 0; m < 16; ++m) {
        // One contiguous 16B A chunk per subtile covers WMMA steps 2m and 2m+1.
        v4f a4[4];
#pragma unroll
        for (int s = 0; s < 4; ++s) {
            const float* xr = X + (size_t)rr[s] * DIM + 8 * m + 4 * half;
            v4f t = *(const v4f*)xr;       // global_load_b128
            a4[s] = t * sc[s];
        }
#pragma unroll
        for (int s = 0; s < 4; ++s) {
            v2f lo = {a4[s].x, a4[s].y};
            acc[s] = __builtin_amdgcn_wmma_f32_16x16x4_f32(
                false, lo, false, bf[2 * m], (short)0, acc[s], false, false);
        }
#pragma unroll
        for (int s = 0; s < 4; ++s) {
            v2f hi = {a4[s].z, a4[s].w};
            acc[s] = __builtin_amdgcn_wmma_f32_16x16x4_f32(
                false, hi, false, bf[2 * m + 1], (short)0, acc[s], false, false);
        }
    }

    float* Hr = H + (size_t)rel * N_NODE * DIM;
#pragma unroll
    for (int s = 0; s < 4; ++s) {
        int rbase = row0 + s * 16 + half * 8;   // C/D: vgpr i -> M = 8*half + i
#pragma unroll
        for (int i = 0; i < 8; ++i) {
            int r = rbase + i;
            if (r < N_NODE) Hr[(size_t)r * DIM + col0 + lm] = acc[s][i];
        }
    }
}

// ---------------------------------------------------------------- edge scatter
// out[dst] += H[rel][src] * degIn[rel][dst]^-1/2   (scale folded per-edge)
// One wave per edge; relations 0,2 -> drug half, 1,3 -> dis half.
__global__ void __launch_bounds__(256)
scatter_edges_kernel(const int* __restrict__ s0, const int* __restrict__ d0,
                     const int* __restrict__ s1, const int* __restrict__ d1,
                     const int* __restrict__ s2, const int* __restrict__ d2,
                     const int* __restrict__ s3, const int* __restrict__ d3,
                     const float* __restrict__ H,
                     const unsigned* __restrict__ degIn,
                     float* __restrict__ out) {
    int w    = threadIdx.x >> 5;
    int lane = threadIdx.x & 31;
    int e    = blockIdx.x * (blockDim.x >> 5) + w;
    int r    = blockIdx.y;
    if (e >= NEDGE) return;
    const int* sp; const int* dp;
    switch (r) {
        case 0:  sp = s0; dp = d0; break;
        case 1:  sp = s1; dp = d1; break;
        case 2:  sp = s2; dp = d2; break;
        default: sp = s3; dp = d3; break;
    }
    int src = sp[e];
    int dst = dp[e];
    float s = rsqrtf(fmaxf((float)degIn[r * N_NODE + dst], 1.0f));
    const float* h = H + ((size_t)r * N_NODE + src) * DIM;
    float*       o = out + ((r & 1) ? (size_t)N_NODE * DIM : (size_t)0)
                         + (size_t)dst * DIM;
#pragma unroll
    for (int j = 0; j < 4; ++j) {
        int c = lane + j * 32;
        unsafeAtomicAdd(&o[c], h[c] * s);     // global_atomic_add_f32
    }
}

// ---------------------------------------------------------------- PReLU
__global__ void prelu_kernel(float* __restrict__ out, const float* __restrict__ a, int n) {
    int i = blockIdx.x * blockDim.x + threadIdx.x;
    if (i >= n) return;
    float v  = out[i];
    out[i]   = (v >= 0.0f) ? v : a[0] * v;
}

// ---------------------------------------------------------------- launch
extern "C" void kernel_launch(void* const* d_in, const int* in_sizes, int n_in,
                              void* d_out, int out_size, void* d_ws, size_t ws_size,
                              hipStream_t stream) {
    const float* x_drug = (const float*)d_in[0];
    const float* x_dis  = (const float*)d_in[1];
    const float* W      = (const float*)d_in[2];   // [4,128,128]
    const float* b      = (const float*)d_in[3];   // [4,128]
    const float* pa     = (const float*)d_in[4];   // [1]
    const int* e_dd_s = (const int*)d_in[5];
    const int* e_dd_d = (const int*)d_in[6];
    const int* e_ds_s = (const int*)d_in[7];
    const int* e_ds_d = (const int*)d_in[8];
    const int* e_sd_s = (const int*)d_in[9];
    const int* e_sd_d = (const int*)d_in[10];
    const int* e_ss_s = (const int*)d_in[11];
    const int* e_ss_d = (const int*)d_in[12];
    float* out = (float*)d_out;

    // workspace layout: H (4*50000*128 f32 = 102.4 MB) | degOut (4*50000 u32) | degIn (4*50000 u32)
    char*     ws     = (char*)d_ws;
    float*    H      = (float*)ws;
    unsigned* degOut = (unsigned*)(ws + (size_t)4 * N_NODE * DIM * sizeof(float));
    unsigned* degIn  = degOut + 4 * N_NODE;

    // 1) zero degree counters
    zero_u32_kernel<<<(8 * N_NODE + 255) / 256, 256, 0, stream>>>(degOut, 8 * N_NODE);

    // 2) degree counts (all 4 relations, grid.y = relation)
    dim3 gcnt((NEDGE + 255) / 256, 4);
    count_degrees_kernel<<<gcnt, 256, 0, stream>>>(e_dd_s, e_dd_d, e_ds_s, e_ds_d,
                                                   e_sd_s, e_sd_d, e_ss_s, e_ss_d,
                                                   degOut, degIn);

    // 3) init d_out with summed biases (accumulation target)
    init_out_bias_kernel<<<(2 * N_NODE * DIM + 255) / 256, 256, 0, stream>>>(b, out);

    // 4) four degree-scaled GEMMs via f32 WMMA
    dim3 ggemm((N_NODE + 63) / 64, DIM / 16, 4);
    gemm_norm_wmma_kernel<<<ggemm, 32, 0, stream>>>(x_drug, x_dis, W, degOut, H);

    // 5) edge scatter with folded in-degree scaling (L2-resident atomics)
    dim3 gsc((NEDGE + 7) / 8, 4);
    scatter_edges_kernel<<<gsc, 256, 0, stream>>>(e_dd_s, e_dd_d, e_ds_s, e_ds_d,
                                                  e_sd_s, e_sd_d, e_ss_s, e_ss_d,
                                                  H, degIn, out);

    // 6) PReLU in place
    prelu_kernel<<<(2 * N_NODE * DIM + 255) / 256, 256, 0, stream>>>(out, pa, 2 * N_NODE * DIM);
}